// GeometricSuperpositionSearch_86500641342134
// MI455X (gfx1250) — compile-verified
//
#include <hip/hip_runtime.h>
#include <cstddef>

#define DEVFN __device__ __forceinline__

typedef __attribute__((ext_vector_type(2)))  float v2f;
typedef __attribute__((ext_vector_type(8)))  float v8f;

#if __has_builtin(__builtin_amdgcn_global_load_async_to_lds_b128)
#define GSS_ASYNC 1
typedef int v4i __attribute__((__vector_size__(16)));
typedef __attribute__((address_space(1))) v4i ga_v4i;   // global b128 unit
typedef __attribute__((address_space(3))) v4i ls_v4i;   // LDS b128 unit
#endif

namespace gss {

constexpr int   Bn = 512, Cn = 256, Hn = 8, An = 5, MIDn = 128, Nn = 4096;
constexpr float EPSf = 1e-6f;

// ---------------------------------------------------------------- helpers ---

DEVFN void load8(const float* __restrict__ p, float* x) {
  float4 u = *(const float4*)p;
  float4 v = *(const float4*)(p + 4);
  x[0]=u.x; x[1]=u.y; x[2]=u.z; x[3]=u.w;
  x[4]=v.x; x[5]=v.y; x[6]=v.z; x[7]=v.w;
}
DEVFN void store8(float* __restrict__ p, const float* x) {
  float4 u = {x[0],x[1],x[2],x[3]};
  float4 v = {x[4],x[5],x[6],x[7]};
  *(float4*)p = u; *(float4*)(p + 4) = v;
}

DEVFN void wait_async0() {
#if __has_builtin(__builtin_amdgcn_s_wait_asynccnt)
  __builtin_amdgcn_s_wait_asynccnt(0);
#else
  asm volatile("s_wait_asynccnt 0x0" ::: "memory");
#endif
}

// Cl(3,0) geometric product, basis order [1,e1,e2,e3,e12,e13,e23,e123].
// Swap-parity for bitmask blades a,b (3 bits): par = a1&b0 ^ a2&b1 ^ a2&b0.
DEVFN void gp8(const float* a, const float* b, float* r) {
  constexpr int MB[8] = {0,1,2,4,3,5,6,7};  // index -> bitmask
  constexpr int BI[8] = {0,1,2,4,3,5,6,7};  // bitmask -> index (involution)
  float acc[8] = {0,0,0,0,0,0,0,0};
#pragma unroll
  for (int i = 0; i < 8; ++i) {
#pragma unroll
    for (int j = 0; j < 8; ++j) {
      const int am = MB[i], bm = MB[j];
      const int a1 = (am >> 1) & 1, a2 = (am >> 2) & 1;
      const int b0 = bm & 1,        b1 = (bm >> 1) & 1;
      const int par = (a1 & b0) ^ (a2 & b1) ^ (a2 & b0);
      acc[BI[am ^ bm]] += (par ? -1.0f : 1.0f) * a[i] * b[j];
    }
  }
#pragma unroll
  for (int k = 0; k < 8; ++k) r[k] = acc[k];
}

// R x reverse(R); reverse flips grades 2,3 (indices 4..7)
DEVFN void sandwich(const float* R, const float* x, float* out) {
  float t[8], Rr[8];
  gp8(R, x, t);
  Rr[0]=R[0]; Rr[1]=R[1]; Rr[2]=R[2]; Rr[3]=R[3];
  Rr[4]=-R[4]; Rr[5]=-R[5]; Rr[6]=-R[6]; Rr[7]=-R[7];
  gp8(t, Rr, out);
}

DEVFN void exp_half_bv(const float* bv, float* R) {  // exp(-0.5 * scatter(bv))
  float bx = -0.5f*bv[0], by = -0.5f*bv[1], bz = -0.5f*bv[2];
  float t  = sqrtf(bx*bx + by*by + bz*bz);
  float sinc = (t < 1e-6f) ? (1.0f - t*t*(1.0f/6.0f)) : (sinf(t)/t);
  R[0]=cosf(t); R[1]=0.f; R[2]=0.f; R[3]=0.f;
  R[4]=sinc*bx; R[5]=sinc*by; R[6]=sinc*bz; R[7]=0.f;
}

DEVFN float gelu_tanh(float x) {  // jax.nn.gelu(approximate=True)
  const float k = 0.7978845608028654f;
  return 0.5f * x * (1.0f + tanhf(k * (x + 0.044715f * x*x*x)));
}

DEVFN float blockReduce256(float v, float* sd) {
  const int t = threadIdx.x;
  sd[t] = v;
  __syncthreads();
#pragma unroll
  for (int s = 128; s > 0; s >>= 1) {
    if (t < s) sd[t] += sd[t + s];
    __syncthreads();
  }
  float r = sd[0];
  __syncthreads();
  return r;
}

// --- exact JAX noise: threefry2x32(key=(0,42)) + uniform(lo,1) + erfinv -----
DEVFN float jax_noise(unsigned flat) {
  const unsigned HALF = 30720u;  // (DEPTH*N*A)/2
  unsigned x0, x1; bool second;
  if (flat < HALF) { x0 = flat;        x1 = flat + HALF; second = false; }
  else             { x0 = flat - HALF; x1 = flat;        second = true;  }
  const unsigned ks0 = 0u, ks1 = 42u, ks2 = 0u ^ 42u ^ 0x1BD11BDAu;
  x0 += ks0; x1 += ks1;
#define TF_R(r) { x0 += x1; x1 = (x1 << (r)) | (x1 >> (32 - (r))); x1 ^= x0; }
  TF_R(13) TF_R(15) TF_R(26) TF_R(6)   x0 += ks1; x1 += ks2 + 1u;
  TF_R(17) TF_R(29) TF_R(16) TF_R(24)  x0 += ks2; x1 += ks0 + 2u;
  TF_R(13) TF_R(15) TF_R(26) TF_R(6)   x0 += ks0; x1 += ks1 + 3u;
  TF_R(17) TF_R(29) TF_R(16) TF_R(24)  x0 += ks1; x1 += ks2 + 4u;
  TF_R(13) TF_R(15) TF_R(26) TF_R(6)   x0 += ks2; x1 += ks0 + 5u;
#undef TF_R
  unsigned bits = second ? x1 : x0;
  float u01 = __uint_as_float((bits >> 9) | 0x3f800000u) - 1.0f;  // [0,1)
  const float lo = -0.9999999403953552f;                          // nextafter(-1,0)
  float u = fmaxf(lo, u01 * (1.0f - lo) + lo);
  return 1.41421356237309515f * erfinvf(u) * 0.3f;
}

// ---------------------------------------------------------------- kernels ---

// precompute rotors: hypothesis [H,8], action [A,8], per-channel [C,8]
__global__ void k_setup(const float* __restrict__ hyp_bv, const float* __restrict__ act_bv,
                        const float* __restrict__ rot_bv,
                        float* __restrict__ R_hyp, float* __restrict__ R_act,
                        float* __restrict__ R_chan) {
  const int t = threadIdx.x;
  if (t < Hn)                       exp_half_bv(hyp_bv + t * 3,        R_hyp  + t * 8);
  else if (t >= 8 && t < 8 + An)    exp_half_bv(act_bv + (t - 8) * 3,  R_act  + (t - 8) * 8);
  else if (t >= 16 && t < 16 + Cn)  exp_half_bv(rot_bv + (t - 16) * 3, R_chan + (t - 16) * 8);
}

// hf[n=b*H+h, c, :] = R_h state[b,c,:] ~R_h
__global__ __launch_bounds__(256) void k_hyp(const float* __restrict__ state,
                                             const float* __restrict__ R_hyp,
                                             float* __restrict__ hf) {
  const int gid = blockIdx.x * 256 + threadIdx.x;  // = n*C + c
  const int n = gid >> 8, c = gid & 255;
  const int b = n >> 3,   h = n & 7;
  float x[8], R[8], out[8];
  load8(state + ((size_t)(b * Cn + c)) * 8, x);
  load8(R_hyp + h * 8, R);
  sandwich(R, x, out);
  store8(hf + (size_t)gid * 8, out);
}

// explore[b] = |tanh(MLP(mean_c sqrt(grade_energy(state)+1e-12)))|
__global__ __launch_bounds__(256) void k_explore(const float* __restrict__ state,
    const float* __restrict__ w1, const float* __restrict__ b1,
    const float* __restrict__ w2, const float* __restrict__ b2,
    float* __restrict__ explore) {
  __shared__ float sd[256];
  const int b = blockIdx.x, c = threadIdx.x;
  float x[8];
  load8(state + ((size_t)(b * Cn + c)) * 8, x);
  float g[4];
  g[0] = x[0]*x[0];
  g[1] = x[1]*x[1] + x[2]*x[2] + x[3]*x[3];
  g[2] = x[4]*x[4] + x[5]*x[5] + x[6]*x[6];
  g[3] = x[7]*x[7];
  float pooled[4];
#pragma unroll
  for (int i = 0; i < 4; ++i)
    pooled[i] = blockReduce256(sqrtf(g[i] + 1e-12f), sd) * (1.0f / 256.0f);
  if (c == 0) {
    float theta = b2[0];
#pragma unroll
    for (int j = 0; j < 32; ++j) {
      float h = b1[j];
#pragma unroll
      for (int i = 0; i < 4; ++i) h += pooled[i] * w1[i * 32 + j];
      theta += fmaxf(h, 0.0f) * w2[j];
    }
    explore[b] = fabsf(tanhf(theta));
  }
}

__global__ void k_init(float* __restrict__ hv) {
  const int t = blockIdx.x * blockDim.x + threadIdx.x;
  if (t < Nn) hv[t] = 0.0f;
}

// invden[n] = 1 / (mean_c sqrt(sum_d x^2 + EPS) + EPS)
__global__ __launch_bounds__(256) void k_denom(const float* __restrict__ hf,
                                               float* __restrict__ invden) {
  __shared__ float sd[256];
  const int n = blockIdx.x, c = threadIdx.x;
  float x[8];
  load8(hf + ((size_t)(n * Cn + c)) * 8, x);
  float s = 0.f;
#pragma unroll
  for (int d = 0; d < 8; ++d) s += x[d] * x[d];
  float tot = blockReduce256(sqrtf(s + EPSf), sd);
  if (c == 0) invden[n] = 1.0f / (tot * (1.0f / 256.0f) + EPSf);
}

// WMMA GEMM: OUT[n,o,d] (+bias[o,d]) = sum_c W[o,c] * (X[n,c,d] * scale(n,c))
// Block: 16 rows (2 n's x 8 blades), 8 waves each own a 16-wide o tile.
// A: staged once to LDS, transposed [row m][c] (pad 260), scale fused at staging.
// B: double-buffered async staging, 32-c chunks (pad 36), copy k+1 overlaps
//    compute of chunk k (GLOBAL_LOAD_ASYNC_TO_LDS_B128 + s_wait_asynccnt).
template<int O, bool SCALE, bool GELU, bool RESID>
__global__ __launch_bounds__(256)
void k_gemm(const float* __restrict__ X, const float* __restrict__ W,
            const float* __restrict__ bias, const float* __restrict__ normw,
            const float* __restrict__ invden, float* __restrict__ OUT) {
  constexpr int AP    = 260;        // padded A c-stride
  constexpr int BP    = 36;         // padded B chunk c-stride
  constexpr int CHUNK = 32;
  constexpr int NCH   = Cn / CHUNK; // 8 chunks
  __shared__ float Ald[16 * AP];    // 16.6 KB
#ifdef GSS_ASYNC
  __shared__ float Bld[2][128 * BP];  // 2 x 18.4 KB
#endif
  const int t    = threadIdx.x;
  const int lane = t & 31;
  const int wave = t >> 5;
  const int n0   = blockIdx.x * 2;
  const int oB   = blockIdx.y * 128;

#ifdef GSS_ASYNC
  // async-copy one 32-c chunk of the 128-row B panel (1024 x b128)
  auto issue_chunk = [&](int buf, int cb) {
#pragma unroll
    for (int i = 0; i < 4; ++i) {
      const int q  = i * 256 + t;
      const int o  = q >> 3;
      const int ci = (q & 7) * 4;
      __builtin_amdgcn_global_load_async_to_lds_b128(
          (ga_v4i*)(W + (size_t)(oB + o) * Cn + cb + ci),
          (ls_v4i*)(&Bld[buf][o * BP + ci]), 0, 0);
    }
  };
  issue_chunk(0, 0);   // chunk 0 copy overlaps the A staging below
#endif

  // ---- stage A: transpose [n,c,d] -> [m=nn*8+d][c], fusing clayernorm scale
  for (int p = t; p < 512; p += 256) {
    const int nn = p >> 8, c = p & 255;
    const int n = n0 + nn;
    float x[8];
    load8(X + ((size_t)(n * Cn + c)) * 8, x);
    float s = 1.0f;
    if (SCALE) s = normw[c] * invden[n];
#pragma unroll
    for (int d = 0; d < 8; ++d) Ald[(nn * 8 + d) * AP + c] = x[d] * s;
  }

  const int hi   = lane >> 4;         // lanes 0-15 -> K{0,1}; 16-31 -> K{2,3}
  const int mrow = lane & 15;
  const int kb   = hi * 2;
  const int oL   = wave * 16 + (lane & 15);
  const float* Ap = Ald + mrow * AP + kb;
  v8f acc = {0.f,0.f,0.f,0.f,0.f,0.f,0.f,0.f};

#ifdef GSS_ASYNC
  for (int k = 0; k < NCH; ++k) {
    const int buf = k & 1;
    wait_async0();        // chunk k copies (issued last iteration) complete
    __syncthreads();      // ...for all waves; also covers A staging at k==0
    if (k + 1 < NCH) issue_chunk(buf ^ 1, (k + 1) * CHUNK);  // overlap copy
    const float* Apc = Ap + k * CHUNK;
    const float* Bp  = &Bld[buf][oL * BP + kb];
#pragma unroll
    for (int cl = 0; cl < CHUNK; cl += 4) {
      float2 av = *(const float2*)(Apc + cl);   // ds_load_b64
      float2 bw = *(const float2*)(Bp + cl);    // ds_load_b64
      v2f a; a[0] = av.x; a[1] = av.y;
      v2f b; b[0] = bw.x; b[1] = bw.y;
      acc = __builtin_amdgcn_wmma_f32_16x16x4_f32(false, a, false, b,
                                                  (short)0, acc, false, false);
    }
    __syncthreads();      // all waves done with buf before it refills at k+2
  }
#else
  __syncthreads();        // A staged
  const float* Bg = W + (size_t)(oB + oL) * Cn + kb;
  for (int cb = 0; cb < Cn; cb += CHUNK) {
    __builtin_prefetch(Bg + cb + CHUNK, 0, 1);  // global_prefetch next chunk
#pragma unroll
    for (int cl = 0; cl < CHUNK; cl += 4) {
      float2 av = *(const float2*)(Ap + cb + cl);   // ds_load_b64
      float2 bw = *(const float2*)(Bg + cb + cl);   // global b64 (L2-hot)
      v2f a; a[0] = av.x; a[1] = av.y;
      v2f b; b[0] = bw.x; b[1] = bw.y;
      acc = __builtin_amdgcn_wmma_f32_16x16x4_f32(false, a, false, b,
                                                  (short)0, acc, false, false);
    }
  }
#endif

  // epilogue: lane (hi, j) holds output multivector (n0+hi, oB+wave*16+j), blades v=0..7
  const int en = n0 + hi;
  const int oC = oB + oL;
  float* op = OUT + ((size_t)en * O + oC) * 8;
  const float* bp = bias + (size_t)oC * 8;
  float r[8];
#pragma unroll
  for (int v = 0; v < 8; ++v) r[v] = acc[v] + bp[v];
  if (GELU) {  // ggelu: gate multivector by gelu of its norm (all blades in-lane)
    float s = 0.f;
#pragma unroll
    for (int v = 0; v < 8; ++v) s += r[v] * r[v];
    float nn = sqrtf(s + EPSf);
    float g  = gelu_tanh(nn) / (nn + EPSf);
#pragma unroll
    for (int v = 0; v < 8; ++v) r[v] *= g;
  }
  if (RESID) {
    float ex[8];
    load8(op, ex);
#pragma unroll
    for (int v = 0; v < 8; ++v) r[v] += ex[v];
  }
  store8(op, r);
}

// heads: value / policy / noise-perturbed argmax.  step 0..2: action; step 3: value only.
__global__ __launch_bounds__(256) void k_heads(const float* __restrict__ h3,
    const float* __restrict__ hf, const float* __restrict__ explore,
    const float* __restrict__ vw, const float* __restrict__ vb,
    const float* __restrict__ pw, const float* __restrict__ pb,
    int* __restrict__ action, int* __restrict__ first_action,
    float* __restrict__ hyp_values, int step) {
  __shared__ float sd[256];
  const int n = blockIdx.x, t = threadIdx.x;
  if (step == 3) {
    float vp = 0.f;
    if (t < MIDn) vp = h3[((size_t)(n * MIDn + t)) * 8] * vw[t];
    float value = blockReduce256(vp, sd) + vb[0];
    if (t == 0) hyp_values[n] += 0.970299f * value;  // DISC^3
    return;
  }
  const float* hp = hf + ((size_t)(n * Cn + t)) * 8;
  float bv = hp[4]*hp[4] + hp[5]*hp[5] + hp[6]*hp[6];
  float bv_e = blockReduce256(bv, sd) * (1.0f / 256.0f);
  float pp[An] = {0.f, 0.f, 0.f, 0.f, 0.f};
  if (t < MIDn) {
    float h[8];
    load8(h3 + ((size_t)(n * MIDn + t)) * 8, h);
#pragma unroll
    for (int gi = 0; gi < 3; ++gi) {
      const float* pr = pw + (size_t)(t * 3 + gi) * An;
      const float hv = h[1 + gi];  // grade-1 blades at 1,2,3
#pragma unroll
      for (int a = 0; a < An; ++a) pp[a] += hv * pr[a];
    }
  }
  float pa[An];
#pragma unroll
  for (int a = 0; a < An; ++a) pa[a] = blockReduce256(pp[a], sd) + pb[a];
  if (t == 0) {
    const float e = explore[n >> 3];
    float best = -3.4e38f; int act = 0;
#pragma unroll
    for (int a = 0; a < An; ++a) {
      float ns = jax_noise((unsigned)((step * Nn + n) * An + a));
      float sc = pa[a] + e * bv_e * ns;
      if (sc > best) { best = sc; act = a; }
    }
    action[n] = act;
    if (step == 0) first_action[n] = act;
  }
}

// dynamics pre-GEMM: ns = R_act hf ~R_act ; t = rotor(clayernorm->ggelu(ns))
__global__ __launch_bounds__(256) void k_dyn_pre(const float* __restrict__ hf,
    const int* __restrict__ action, const float* __restrict__ R_act,
    const float* __restrict__ R_chan, const float* __restrict__ dnw,
    float* __restrict__ ns_out, float* __restrict__ t_out) {
  __shared__ float sd[256];
  const int n = blockIdx.x, c = threadIdx.x;
  float R[8], x[8], ns[8];
  load8(R_act + action[n] * 8, R);
  load8(hf + ((size_t)(n * Cn + c)) * 8, x);
  sandwich(R, x, ns);
  store8(ns_out + ((size_t)(n * Cn + c)) * 8, ns);
  float s = 0.f;
#pragma unroll
  for (int d = 0; d < 8; ++d) s += ns[d] * ns[d];
  float denom = blockReduce256(sqrtf(s + EPSf), sd) * (1.0f / 256.0f) + EPSf;
  const float w = dnw[c] / denom;
  float y[8];
#pragma unroll
  for (int d = 0; d < 8; ++d) y[d] = ns[d] * w;
  float s2 = 0.f;
#pragma unroll
  for (int d = 0; d < 8; ++d) s2 += y[d] * y[d];
  float nn = sqrtf(s2 + EPSf);
  float g  = gelu_tanh(nn) / (nn + EPSf);
#pragma unroll
  for (int d = 0; d < 8; ++d) y[d] *= g;
  float Rc[8], t8[8];
  load8(R_chan + c * 8, Rc);
  sandwich(Rc, y, t8);
  store8(t_out + ((size_t)(n * Cn + c)) * 8, t8);
}

__global__ __launch_bounds__(256) void k_reward(const float* __restrict__ ns,
    const float* __restrict__ rw, const float* __restrict__ rb,
    float coef, float* __restrict__ hyp_values) {
  __shared__ float sd[256];
  const int n = blockIdx.x, c = threadIdx.x;
  float p = ns[((size_t)(n * Cn + c)) * 8] * rw[c];
  float s = blockReduce256(p, sd);
  if (c == 0) hyp_values[n] += coef * (s + rb[0]);
}

__global__ void k_final(const float* __restrict__ hv, const int* __restrict__ fa,
                        float* __restrict__ out) {
  const int b = blockIdx.x * blockDim.x + threadIdx.x;
  if (b >= Bn) return;
  float v[Hn], m = -3.4e38f;
#pragma unroll
  for (int h = 0; h < Hn; ++h) { v[h] = hv[b * Hn + h]; m = fmaxf(m, v[h]); }
  float w[Hn], s = 0.f;
#pragma unroll
  for (int h = 0; h < Hn; ++h) { w[h] = expf(v[h] - m); s += w[h]; }
  float ap[An] = {0.f, 0.f, 0.f, 0.f, 0.f};
#pragma unroll
  for (int h = 0; h < Hn; ++h) ap[fa[b * Hn + h]] += w[h] / s;
  float s2 = 0.f;
#pragma unroll
  for (int a = 0; a < An; ++a) s2 += ap[a];
  s2 = fmaxf(s2, 1e-8f);
#pragma unroll
  for (int a = 0; a < An; ++a) out[b * An + a] = ap[a] / s2;
}

}  // namespace gss

// ------------------------------------------------------------------- host ---

extern "C" void kernel_launch(void* const* d_in, const int* in_sizes, int n_in,
                              void* d_out, int out_size, void* d_ws, size_t ws_size,
                              hipStream_t stream) {
  using namespace gss;
  (void)in_sizes; (void)n_in; (void)out_size; (void)ws_size;
  const float* state    = (const float*)d_in[0];
  const float* hyp_bv   = (const float*)d_in[1];
  const float* sp_w1    = (const float*)d_in[2];
  const float* sp_b1    = (const float*)d_in[3];
  const float* sp_w2    = (const float*)d_in[4];
  const float* sp_b2    = (const float*)d_in[5];
  const float* p_norm_w = (const float*)d_in[6];
  const float* p_lin_w  = (const float*)d_in[7];
  const float* p_lin_b  = (const float*)d_in[8];
  const float* p_vw     = (const float*)d_in[9];
  const float* p_vb     = (const float*)d_in[10];
  const float* p_pw     = (const float*)d_in[11];
  const float* p_pb     = (const float*)d_in[12];
  const float* act_bv   = (const float*)d_in[13];
  const float* d_norm_w = (const float*)d_in[14];
  const float* d_rotor  = (const float*)d_in[15];
  const float* d_lin_w  = (const float*)d_in[16];
  const float* d_lin_b  = (const float*)d_in[17];
  const float* d_rw     = (const float*)d_in[18];
  const float* d_rb     = (const float*)d_in[19];

  float* ws = (float*)d_ws;
  const size_t NC8 = (size_t)Nn * Cn * 8;           // 8,388,608 floats
  float* hfA = ws;
  float* hfB = ws + NC8;
  float* tmp = ws + 2 * NC8;                        // predict h3 / dynamics t
  float* sm  = ws + 3 * NC8;                        // ~100.7 MB used total
  float* R_hyp  = sm;                               // 64
  float* R_act  = sm + 64;                          // 40 (padded)
  float* R_chan = sm + 128;                         // 2048
  float* invden = sm + 2176;                        // 4096
  float* expl   = sm + 6272;                        // 512
  float* hv     = sm + 6784;                        // 4096
  int*   act    = (int*)(sm + 10880);               // 4096
  int*   fact   = (int*)(sm + 14976);               // 4096

  k_setup<<<1, 512, 0, stream>>>(hyp_bv, act_bv, d_rotor, R_hyp, R_act, R_chan);
  k_hyp<<<Nn, 256, 0, stream>>>(state, R_hyp, hfA);
  k_explore<<<Bn, 256, 0, stream>>>(state, sp_w1, sp_b1, sp_w2, sp_b2, expl);
  k_init<<<16, 256, 0, stream>>>(hv);

  float* cur = hfA; float* nxt = hfB;
  const float disc[3] = {1.0f, 0.99f, 0.9801f};
  for (int d = 0; d < 3; ++d) {
    k_denom<<<Nn, 256, 0, stream>>>(cur, invden);
    k_gemm<MIDn, true, true, false><<<dim3(Nn / 2, 1), 256, 0, stream>>>(
        cur, p_lin_w, p_lin_b, p_norm_w, invden, tmp);
    k_heads<<<Nn, 256, 0, stream>>>(tmp, cur, expl, p_vw, p_vb, p_pw, p_pb,
                                    act, fact, hv, d);
    k_dyn_pre<<<Nn, 256, 0, stream>>>(cur, act, R_act, R_chan, d_norm_w, nxt, tmp);
    k_gemm<Cn, false, false, true><<<dim3(Nn / 2, 2), 256, 0, stream>>>(
        tmp, d_lin_w, d_lin_b, nullptr, nullptr, nxt);
    k_reward<<<Nn, 256, 0, stream>>>(nxt, d_rw, d_rb, disc[d], hv);
    float* t2 = cur; cur = nxt; nxt = t2;
  }
  k_denom<<<Nn, 256, 0, stream>>>(cur, invden);
  k_gemm<MIDn, true, true, false><<<dim3(Nn / 2, 1), 256, 0, stream>>>(
      cur, p_lin_w, p_lin_b, p_norm_w, invden, tmp);
  k_heads<<<Nn, 256, 0, stream>>>(tmp, cur, expl, p_vw, p_vb, p_pw, p_pb,
                                  act, fact, hv, 3);
  k_final<<<2, 256, 0, stream>>>(hv, fact, (float*)d_out);
}